// PermutoEncoding_44616120270959
// MI455X (gfx1250) — compile-verified
//
#include <hip/hip_runtime.h>

// Permutohedral hash-grid encoding, specialized for
// d=3 (dp1=4), L=24 levels, C=2^18 capacity, F=2 features, N=524288 points.
//
// One thread per (point, level). level = tid % 24 so the (N, L*F) output is
// written as perfectly coalesced float2 stores (non-temporal: the 100MB output
// stream must not evict the 50MB of hash tables resident in the 192MB L2).
// Gathers use TH_LOAD_NT_RT: no WGP$ allocation (random over 50MB => ~0% L0
// hit rate, would only thrash the near cache), regular temporal in L2.

#define PE_POS_DIM   3
#define PE_DP1       4
#define PE_CAP       262144        // 2^18
#define PE_CAP_MASK  (PE_CAP - 1)
#define PE_LEVELS    24
#define PE_FEAT      2
#define PE_NPTS      524288
#define PE_HASH_MUL  2531011u

typedef float pe_f2 __attribute__((ext_vector_type(2)));

__global__ __launch_bounds__(256) void permuto_encode_kernel(
    const float* __restrict__ positions,   // (N, 3)
    const float* __restrict__ lattice,     // (L, C, 2)
    const float* __restrict__ rshift,      // (L, 3)
    const float* __restrict__ anneal,      // (L,)
    const float* __restrict__ scale,       // (L,)
    float* __restrict__ out)               // (N, L*2)
{
    // ---- stage tiny per-level constants in LDS (120 floats total) ----
    __shared__ float sh_shift[PE_LEVELS * PE_POS_DIM];
    __shared__ float sh_scale[PE_LEVELS];
    __shared__ float sh_anneal[PE_LEVELS];
    if (threadIdx.x < PE_LEVELS * PE_POS_DIM)
        sh_shift[threadIdx.x] = rshift[threadIdx.x];
    if (threadIdx.x < PE_LEVELS) {
        sh_scale[threadIdx.x]  = scale[threadIdx.x];
        sh_anneal[threadIdx.x] = anneal[threadIdx.x];
    }
    __syncthreads();

    const unsigned tid = blockIdx.x * blockDim.x + threadIdx.x;
    if (tid >= (unsigned)PE_NPTS * PE_LEVELS) return;
    const unsigned l = tid % PE_LEVELS;
    const unsigned n = tid / PE_LEVELS;

    // Warm L2 for the streamed position lines a few blocks ahead
    // (gfx1250 global_prefetch_b8; OOB speculative prefetch is dropped).
    if (threadIdx.x == 0) {
        size_t pf_pt = (size_t)(blockIdx.x + 16) * 256u / PE_LEVELS;
        __builtin_prefetch(positions + pf_pt * 3, 0, 1);
    }

    const float px = positions[(size_t)n * 3 + 0];
    const float py = positions[(size_t)n * 3 + 1];
    const float pz = positions[(size_t)n * 3 + 2];

    const float sc = sh_scale[l];
    // scale / sqrt((i+1)(i+2)) for i = 0,1,2
    const float cf0 = (px + sh_shift[l * 3 + 0]) * (sc * 0.70710678118654752f); // 1/sqrt(2)
    const float cf1 = (py + sh_shift[l * 3 + 1]) * (sc * 0.40824829046386302f); // 1/sqrt(6)
    const float cf2 = (pz + sh_shift[l * 3 + 2]) * (sc * 0.28867513459481288f); // 1/sqrt(12)

    // elevate onto hyperplane sum(e)=0
    float e[PE_DP1];
    e[0] = cf0 + cf1 + cf2;
    e[1] = cf1 + cf2 - cf0;
    e[2] = cf2 - 2.0f * cf1;
    e[3] = -3.0f * cf2;

    // nearest multiple-of-4 lattice point per coordinate
    float rem0[PE_DP1];
    float sum4 = 0.0f;
    #pragma unroll
    for (int i = 0; i < PE_DP1; ++i) {
        const float v  = e[i] * 0.25f;
        const float up = ceilf(v) * 4.0f;
        const float dn = floorf(v) * 4.0f;
        rem0[i] = (up - e[i] < e[i] - dn) ? up : dn;
        sum4 += rem0[i];
    }
    const int sum_val = (int)rintf(sum4 * 0.25f);

    // rank of residuals (pairwise: exactly one of each pair increments)
    int ranks[PE_DP1] = {0, 0, 0, 0};
    #pragma unroll
    for (int i = 0; i < PE_DP1; ++i) {
        #pragma unroll
        for (int j = i + 1; j < PE_DP1; ++j) {
            const float di = e[i] - rem0[i];
            const float dj = e[j] - rem0[j];
            if (di < dj) ranks[i]++; else ranks[j]++;
        }
    }

    // wrap fixup + residual deltas + integer lattice coords
    float delta[PE_DP1];
    int   remi[PE_DP1];
    #pragma unroll
    for (int i = 0; i < PE_DP1; ++i) {
        int   r  = ranks[i] + sum_val;
        float rm = rem0[i];
        if (r < 0)      { r += PE_DP1; rm += (float)PE_DP1; }
        else if (r > 3) { r -= PE_DP1; rm -= (float)PE_DP1; }
        ranks[i] = r;
        delta[i] = (e[i] - rm) * 0.25f;
        remi[i]  = (int)rintf(rm);
    }

    // barycentric weights (branch-free selects, stays in VGPRs)
    float b0 = 0.f, b1 = 0.f, b2 = 0.f, b3 = 0.f, b4 = 0.f;
    #pragma unroll
    for (int i = 0; i < PE_DP1; ++i) {
        const int   t = 3 - ranks[i];
        const float d = delta[i];
        b0 += (t == 0) ? d : 0.f;
        b1 += (t == 1) ? d : ((t == 0) ? -d : 0.f);
        b2 += (t == 2) ? d : ((t == 1) ? -d : 0.f);
        b3 += (t == 3) ? d : ((t == 2) ? -d : 0.f);
        b4 += (t == 3) ? -d : 0.f;
    }
    b0 += 1.0f + b4;

    // hash the 4 simplex corners
    const float* __restrict__ ltab = lattice + (size_t)l * ((size_t)PE_CAP * PE_FEAT);
    unsigned ix[PE_DP1];
    #pragma unroll
    for (int r = 0; r < PE_DP1; ++r) {
        unsigned h = 0u;
        #pragma unroll
        for (int i = 0; i < PE_POS_DIM; ++i) {
            const int key = remi[i] + r - ((ranks[i] > 3 - r) ? PE_DP1 : 0);
            h = (h + (unsigned)key) * PE_HASH_MUL;
        }
        ix[r] = h & PE_CAP_MASK;
    }

    // 4 overlapped float2 gathers with split temporal hint:
    // NT in WGP$ (near), RT in L2 (far). One wait for all four.
    const float* a0 = ltab + (size_t)ix[0] * PE_FEAT;
    const float* a1 = ltab + (size_t)ix[1] * PE_FEAT;
    const float* a2 = ltab + (size_t)ix[2] * PE_FEAT;
    const float* a3 = ltab + (size_t)ix[3] * PE_FEAT;
    pe_f2 g0, g1, g2, g3;
    asm volatile(
        "global_load_b64 %0, %4, off th:TH_LOAD_NT_RT\n\t"
        "global_load_b64 %1, %5, off th:TH_LOAD_NT_RT\n\t"
        "global_load_b64 %2, %6, off th:TH_LOAD_NT_RT\n\t"
        "global_load_b64 %3, %7, off th:TH_LOAD_NT_RT\n\t"
        "s_wait_loadcnt 0x0"
        : "=&v"(g0), "=&v"(g1), "=&v"(g2), "=&v"(g3)
        : "v"(a0), "v"(a1), "v"(a2), "v"(a3)
        : "memory");

    float o0, o1;
    o0 = b0 * g0.x;            o1 = b0 * g0.y;
    o0 = fmaf(b1, g1.x, o0);   o1 = fmaf(b1, g1.y, o1);
    o0 = fmaf(b2, g2.x, o0);   o1 = fmaf(b2, g2.y, o1);
    o0 = fmaf(b3, g3.x, o0);   o1 = fmaf(b3, g3.y, o1);

    const float aw = sh_anneal[l];
    pe_f2 res;
    res.x = o0 * aw;
    res.y = o1 * aw;
    // Non-temporal store: keep the 100MB output stream out of L2 so the
    // hash tables (50MB) stay resident.
    __builtin_nontemporal_store(res,
        (pe_f2*)(out + (size_t)n * (PE_LEVELS * PE_FEAT) + (size_t)l * PE_FEAT));
}

extern "C" void kernel_launch(void* const* d_in, const int* in_sizes, int n_in,
                              void* d_out, int out_size, void* d_ws, size_t ws_size,
                              hipStream_t stream) {
    (void)in_sizes; (void)n_in; (void)d_ws; (void)ws_size; (void)out_size;
    const float* positions = (const float*)d_in[0];   // (N, 3)
    const float* lattice   = (const float*)d_in[1];   // (L, C, 2)
    const float* rshift    = (const float*)d_in[2];   // (L, 3)
    const float* anneal    = (const float*)d_in[3];   // (L,)
    const float* scale     = (const float*)d_in[4];   // (L,)
    float* out = (float*)d_out;                       // (N, L*2)

    const unsigned total  = (unsigned)PE_NPTS * PE_LEVELS;   // 12,582,912
    const unsigned block  = 256;                              // 8 wave32 waves
    const unsigned nblock = (total + block - 1) / block;      // 49152 exactly

    permuto_encode_kernel<<<nblock, block, 0, stream>>>(
        positions, lattice, rshift, anneal, scale, out);
}